// AttentionDecoderLSTM_6614249636285
// MI455X (gfx1250) — compile-verified
//
#include <hip/hip_runtime.h>
#include <math.h>

#define BB 32
#define TT 64
#define SS 128
#define HH 512
#define VV 32000
#define LL 2
#define G4 (4 * HH)

typedef __attribute__((ext_vector_type(2))) float v2f;
typedef __attribute__((ext_vector_type(8))) float v8f;

// ---------------------------------------------------------------------------
// fp32 WMMA: D(16x16) = A(16x4) * B(4x16) + C   (V_WMMA_F32_16X16X4_F32)
// ---------------------------------------------------------------------------
__device__ __forceinline__ v8f wmma_f32_16x16x4(v2f a, v2f b, v8f c) {
  return __builtin_amdgcn_wmma_f32_16x16x4_f32(
      /*neg_a=*/false, a, /*neg_b=*/false, b,
      /*c_mod=*/(short)0, c, /*reuse_a=*/false, /*reuse_b=*/false);
}

__device__ __forceinline__ float sigmoidf_(float x) {
  return 1.0f / (1.0f + __expf(-x));
}

// ---------------------------------------------------------------------------
// Workspace layout (float offsets). First 64 bytes = barrier ints.
// ---------------------------------------------------------------------------
#define WS_H    16                          // h double buffer [2][L][B][H]
#define WS_C    (WS_H + 2 * LL * BB * HH)   // c               [L][B][H]
#define WS_TOPH (WS_C + LL * BB * HH)       // top-layer h     [T][B][H]
#define WS_OATT (WS_TOPH + TT * BB * HH)    // out_att         [B*T][H], r=b*T+t

// ---------------------------------------------------------------------------
// Device-scope generation barrier (32 persistent workgroups).
// ---------------------------------------------------------------------------
__device__ __forceinline__ void grid_sync(int* cnt, int* gen, int nblk) {
  __syncthreads();
  if (threadIdx.x == 0) {
    __threadfence();  // publish prior global stores
    int g = __hip_atomic_load(gen, __ATOMIC_ACQUIRE, __HIP_MEMORY_SCOPE_AGENT);
    int a = __hip_atomic_fetch_add(cnt, 1, __ATOMIC_ACQ_REL, __HIP_MEMORY_SCOPE_AGENT);
    if (a == nblk - 1) {
      __hip_atomic_store(cnt, 0, __ATOMIC_RELAXED, __HIP_MEMORY_SCOPE_AGENT);
      __hip_atomic_fetch_add(gen, 1, __ATOMIC_ACQ_REL, __HIP_MEMORY_SCOPE_AGENT);
    } else {
      while (__hip_atomic_load(gen, __ATOMIC_ACQUIRE, __HIP_MEMORY_SCOPE_AGENT) == g) {
        __builtin_amdgcn_s_sleep(2);
      }
    }
  }
  __syncthreads();
}

// ---------------------------------------------------------------------------
// Kernel 1: persistent LSTM recurrence over all T steps.
// 32 WGs; WG g owns hidden units j in [16g, 16g+16) for both layers.
// Per layer-step each of the 8 waves computes one 16x16 gate tile:
//   wave = (mi | gi<<1): rows b in [16mi,16mi+16), gate gi, cols = j slice.
// ---------------------------------------------------------------------------
__global__ __launch_bounds__(256) void lstm_recurrence_kernel(
    const int* __restrict__ tgt,
    const float* __restrict__ h0, const float* __restrict__ c0,
    const float* __restrict__ emb,
    const float* __restrict__ W_ih, const float* __restrict__ W_hh,
    const float* __restrict__ b_ih, const float* __restrict__ b_hh,
    float* __restrict__ ws, float* __restrict__ out_hc, int nblk) {
  int* cnt = (int*)ws;
  int* gen = cnt + 1;
  float* hb0  = ws + WS_H;     // [2][L][B][H] double buffer
  float* cbuf = ws + WS_C;     // [L][B][H]
  float* toph = ws + WS_TOPH;  // [T][B][H]

  const int tid   = threadIdx.x;
  const int lane  = tid & 31;
  const int wave  = tid >> 5;
  const int jbase = blockIdx.x * 16;
  const int mi    = wave & 1;   // row tile (b base = 16*mi)
  const int gi    = wave >> 1;  // gate index 0..3 (i,f,g,o)
  const int l15   = lane & 15;
  const int khalf = lane >> 4;  // 0|1
  const int koff  = khalf * 2;  // K pair base for A/B fragments

  __shared__ float lds_g[4][BB][16];  // [gate][b][j]  8 KB
  __shared__ int   lds_tok[BB];

  // init h (buffer 0) / c slices from h0, c0
  for (int idx = tid; idx < LL * BB * 16; idx += 256) {
    int l = idx / (BB * 16);
    int b = (idx >> 4) % BB;
    int j = idx & 15;
    int off = (l * BB + b) * HH + jbase + j;
    hb0[off]  = h0[off];
    cbuf[off] = c0[off];
  }
  grid_sync(cnt, gen, nblk);

  const int bA = mi * 16 + l15;           // batch row for A fragments
  const int nW = gi * HH + jbase + l15;   // gate column / W row for B fragments

  for (int t = 0; t < TT; ++t) {
    if (tid < BB) lds_tok[tid] = tgt[tid * TT + t];
    __syncthreads();

    const int p = t & 1;
    const float* hread = hb0 + p * (LL * BB * HH);
    float* hwrite      = hb0 + (1 - p) * (LL * BB * HH);

    for (int l = 0; l < LL; ++l) {
      const float* xrow = (l == 0) ? (emb + (size_t)lds_tok[bA] * HH)
                                   : (hwrite + (size_t)bA * HH);  // layer0 h(t)
      const float* hrow = hread + (size_t)(l * BB + bA) * HH;
      const float* Wi = W_ih + (size_t)l * G4 * HH + (size_t)nW * HH;
      const float* Wh = W_hh + (size_t)l * G4 * HH + (size_t)nW * HH;

      v8f acc = {0.f, 0.f, 0.f, 0.f, 0.f, 0.f, 0.f, 0.f};
#pragma unroll 8
      for (int k = 0; k < HH; k += 4) {  // x @ W_ih.T
        v2f a = *(const v2f*)(xrow + k + koff);
        v2f w = *(const v2f*)(Wi + k + koff);
        acc = wmma_f32_16x16x4(a, w, acc);
      }
#pragma unroll 8
      for (int k = 0; k < HH; k += 4) {  // h @ W_hh.T
        v2f a = *(const v2f*)(hrow + k + koff);
        v2f w = *(const v2f*)(Wh + k + koff);
        acc = wmma_f32_16x16x4(a, w, acc);
      }
      const int nb = l * G4 + nW;
      const float bias = b_ih[nb] + b_hh[nb];
#pragma unroll
      for (int r = 0; r < 8; ++r)  // D: M = r + 8*khalf (+16*mi), N = l15
        lds_g[gi][mi * 16 + khalf * 8 + r][l15] = acc[r] + bias;
      __syncthreads();

      // pointwise LSTM update for this WG's hidden slice
      for (int idx = tid; idx < BB * 16; idx += 256) {
        int b = idx >> 4, j = idx & 15;
        float ig = sigmoidf_(lds_g[0][b][j]);
        float fg = sigmoidf_(lds_g[1][b][j]);
        float gg = tanhf(lds_g[2][b][j]);
        float og = sigmoidf_(lds_g[3][b][j]);
        int off = (l * BB + b) * HH + jbase + j;
        float cn = fg * cbuf[off] + ig * gg;
        float hn = og * tanhf(cn);
        cbuf[off]   = cn;
        hwrite[off] = hn;
        if (l == LL - 1) toph[(t * BB + b) * HH + jbase + j] = hn;
      }
      grid_sync(cnt, gen, nblk);  // publish h before anyone consumes it
    }
  }

  // final hidden state -> d_out tail (h then c), [L][B][H] each
  const int fb = 1 - ((TT - 1) & 1);
  const float* hfin = hb0 + fb * (LL * BB * HH);
  for (int idx = tid; idx < LL * BB * 16; idx += 256) {
    int l = idx / (BB * 16);
    int b = (idx >> 4) % BB;
    int j = idx & 15;
    int off = (l * BB + b) * HH + jbase + j;
    out_hc[off] = hfin[off];
    out_hc[LL * BB * HH + off] = cbuf[off];
  }
}

// ---------------------------------------------------------------------------
// Kernel 2: Luong attention + Wc projection, one block per (b,t).
// ---------------------------------------------------------------------------
__global__ __launch_bounds__(128) void attention_kernel(
    const float* __restrict__ enc, const float* __restrict__ Wc,
    const float* __restrict__ bc, const float* __restrict__ toph,
    float* __restrict__ out_att) {
  const int r = blockIdx.x;  // r = b*T + t
  const int b = r / TT, t = r % TT;
  const int tid = threadIdx.x;

  __shared__ float sx[2 * HH];  // [ h | context ]
  __shared__ float ssc[SS];
  __shared__ float red[SS];

  const float* hrow = toph + (size_t)(t * BB + b) * HH;
  const float* eb   = enc + (size_t)b * SS * HH;

  for (int k = tid; k < HH; k += 128) sx[k] = hrow[k];
  __syncthreads();

  // scores[s] = h . enc[b,s,:]   (128 threads == S)
  float sc = 0.f;
  {
    const float* er = eb + (size_t)tid * HH;
    for (int k = 0; k < HH; ++k) sc = fmaf(sx[k], er[k], sc);
  }
  red[tid] = sc;
  __syncthreads();
  for (int o = 64; o > 0; o >>= 1) {
    if (tid < o) red[tid] = fmaxf(red[tid], red[tid + o]);
    __syncthreads();
  }
  const float mx = red[0];
  __syncthreads();
  const float e = __expf(sc - mx);
  red[tid] = e;
  __syncthreads();
  for (int o = 64; o > 0; o >>= 1) {
    if (tid < o) red[tid] += red[tid + o];
    __syncthreads();
  }
  const float inv = 1.f / red[0];
  __syncthreads();
  ssc[tid] = e * inv;
  __syncthreads();

  // context[h] = sum_s attn[s] * enc[b,s,h]
  for (int hh = tid; hh < HH; hh += 128) {
    float cx = 0.f;
    for (int s = 0; s < SS; ++s) cx = fmaf(ssc[s], eb[(size_t)s * HH + hh], cx);
    sx[HH + hh] = cx;
  }
  __syncthreads();

  // out_att[j] = tanh( [h|ctx] . Wc[j,:] + bc[j] )
  for (int j = tid; j < HH; j += 128) {
    const float* wr = Wc + (size_t)j * (2 * HH);
    float a = bc[j];
    for (int k = 0; k < 2 * HH; ++k) a = fmaf(sx[k], wr[k], a);
    out_att[(size_t)r * HH + j] = tanhf(a);
  }
}

// ---------------------------------------------------------------------------
// Kernel 3: vocab projection  preds[r,v] = out_att[r,:] . Wo[v,:] + bo[v]
// M=2048, N=32000, K=512. Each WG: 32x64 tile; 8 waves x one 16x16 WMMA tile.
// ---------------------------------------------------------------------------
__global__ __launch_bounds__(256) void vocab_gemm_kernel(
    const float* __restrict__ A, const float* __restrict__ Wo,
    const float* __restrict__ bo, float* __restrict__ preds) {
  const int lane = threadIdx.x & 31;
  const int wave = threadIdx.x >> 5;
  const int mi = wave & 1, ni = wave >> 1;
  const int m0 = blockIdx.x * 32 + mi * 16;
  const int n0 = blockIdx.y * 64 + ni * 16;
  const int l15 = lane & 15;
  const int koff = (lane >> 4) * 2;

  const float* arow = A + (size_t)(m0 + l15) * HH;
  const float* brow = Wo + (size_t)(n0 + l15) * HH;
  __builtin_prefetch(arow, 0, 3);
  __builtin_prefetch(brow, 0, 1);

  v8f acc = {0.f, 0.f, 0.f, 0.f, 0.f, 0.f, 0.f, 0.f};
#pragma unroll 8
  for (int k = 0; k < HH; k += 4) {
    v2f a = *(const v2f*)(arow + k + koff);
    v2f w = *(const v2f*)(brow + k + koff);
    acc = wmma_f32_16x16x4(a, w, acc);
  }
  const float bias = bo[n0 + l15];
  const int mr = m0 + (lane >> 4) * 8;
  const int nc = n0 + l15;
#pragma unroll
  for (int r = 0; r < 8; ++r)
    preds[(size_t)(mr + r) * VV + nc] = acc[r] + bias;
}

// ---------------------------------------------------------------------------
extern "C" void kernel_launch(void* const* d_in, const int* in_sizes, int n_in,
                              void* d_out, int out_size, void* d_ws,
                              size_t ws_size, hipStream_t stream) {
  const int*   tgt  = (const int*)d_in[0];
  const float* h0   = (const float*)d_in[1];
  const float* c0   = (const float*)d_in[2];
  const float* enc  = (const float*)d_in[3];
  const float* emb  = (const float*)d_in[4];
  const float* W_ih = (const float*)d_in[5];
  const float* W_hh = (const float*)d_in[6];
  const float* b_ih = (const float*)d_in[7];
  const float* b_hh = (const float*)d_in[8];
  const float* Wc   = (const float*)d_in[9];
  const float* bc   = (const float*)d_in[10];
  const float* Wo   = (const float*)d_in[11];
  const float* bo   = (const float*)d_in[12];

  float* preds  = (float*)d_out;                       // [B][T][V]
  float* out_hc = preds + (size_t)BB * TT * VV;        // h[L][B][H], c[L][B][H]
  float* ws     = (float*)d_ws;

  // reset barrier state (graph-capture legal)
  hipMemsetAsync(d_ws, 0, 256, stream);

  lstm_recurrence_kernel<<<32, 256, 0, stream>>>(
      tgt, h0, c0, emb, W_ih, W_hh, b_ih, b_hh, ws, out_hc, 32);
  attention_kernel<<<BB * TT, 128, 0, stream>>>(
      enc, Wc, bc, ws + WS_TOPH, ws + WS_OATT);
  vocab_gemm_kernel<<<dim3((BB * TT) / 32, VV / 64), 256, 0, stream>>>(
      ws + WS_OATT, Wo, bo, preds);
}